// MambaLayer_40819369181529
// MI455X (gfx1250) — compile-verified
//
#include <hip/hip_runtime.h>
#include <hip/hip_bf16.h>
#include <math.h>

// ---------------- problem constants ----------------
#define B_    4
#define S_    2048
#define D_    256
#define DI_   512
#define DS_   16
#define KC_   4      // conv taps
#define DTR_  16
#define W_    256
#define NROW  (B_ * S_)       // 8192 rows for all row-major GEMMs
#define NDBC  (DTR_ + 2*DS_)  // 48
#define NC_   16              // scan chunks
#define CS_   (S_ / NC_)      // 128 steps per chunk

typedef __attribute__((ext_vector_type(2))) float v2f;
typedef __attribute__((ext_vector_type(8))) float v8f;

// ---------------- fp32 WMMA GEMM core --------------------------------------
// One wave computes a 16 x (16*NT) strip of C = A(MxK) * B(KxN), fp32 exact.
// A-fragment layout (ISA 7.12.2, 32-bit A 16x4): lanes 0-15 hold K=k,k+1 for
// row M=lane; lanes 16-31 hold K=k+2,k+3. B/C/D are row-striped across lanes.
template <int NT>
__device__ __forceinline__ void wmma_strip(const float* __restrict__ A,
                                           const float* __restrict__ Bm,
                                           int lda, int ldb, int K,
                                           int row0, int col0, v8f* acc) {
  const int lane = threadIdx.x & 31;
  const int half = lane >> 4;   // 0: K even pair, 1: K odd pair
  const int l16  = lane & 15;
  const float* ap  = A  + (size_t)(row0 + l16) * lda + half * 2;
  const float* bp0 = Bm + (size_t)(half * 2) * ldb + col0 + l16;
  for (int k = 0; k < K; k += 4) {
    v2f a;
    a.x = ap[k];
    a.y = ap[k + 1];
    const float* bp = bp0 + (size_t)k * ldb;
#pragma unroll
    for (int t = 0; t < NT; ++t) {
      v2f b;
      b.x = bp[t * 16];
      b.y = bp[t * 16 + ldb];
      acc[t] = __builtin_amdgcn_wmma_f32_16x16x4_f32(
          false, a, false, b, (short)0, acc[t], false, false);
    }
  }
}

// C/D layout: VGPR r -> row (row0 + r + half*8), col (col0 + t*16 + l16)
#define TILE_STORE_IDX()                        \
  const int lane = threadIdx.x & 31;            \
  const int half = lane >> 4;                   \
  const int l16  = lane & 15;

__device__ __forceinline__ float silu_(float v) { return v / (1.f + expf(-v)); }
__device__ __forceinline__ float elu_(float v)  { return v > 0.f ? v : expm1f(v); }

// ---------------- 1) xz = x @ in_proj_w  (8192x256 * 256x1024) -------------
__global__ void k_in_proj(const float* __restrict__ x,
                          const float* __restrict__ w,
                          float* __restrict__ xz) {
  v8f acc[4] = {};
  const int row0 = blockIdx.y * 16;
  const int col0 = blockIdx.x * 64;
  wmma_strip<4>(x, w, D_, 2 * DI_, D_, row0, col0, acc);
  TILE_STORE_IDX();
#pragma unroll
  for (int t = 0; t < 4; ++t)
#pragma unroll
    for (int r = 0; r < 8; ++r)
      xz[(size_t)(row0 + half * 8 + r) * (2 * DI_) + col0 + t * 16 + l16] =
          acc[t][r];
}

// ---------------- 2) depthwise causal conv + bias + SiLU -------------------
__global__ void k_conv_silu(const float* __restrict__ xz,
                            const float* __restrict__ cw,
                            const float* __restrict__ cb,
                            float* __restrict__ xcv) {
  const int idx = blockIdx.x * blockDim.x + threadIdx.x;
  if (idx >= NROW * DI_) return;
  const int d   = idx % DI_;
  const int row = idx / DI_;    // b*S + s
  const int s   = row % S_;
  float acc = cb[d];
#pragma unroll
  for (int j = 0; j < KC_; ++j) {
    const int sp = s - (KC_ - 1) + j;           // causal pad within batch
    if (sp >= 0)
      acc += xz[(size_t)(row - (KC_ - 1) + j) * (2 * DI_) + d] * cw[d * KC_ + j];
  }
  xcv[idx] = silu_(acc);
}

// ---------------- 3) dbc = xcv @ x_proj_w (N=48) ---------------------------
__global__ void k_x_proj(const float* __restrict__ xcv,
                         const float* __restrict__ w,
                         float* __restrict__ dbc) {
  v8f acc[3] = {};
  const int row0 = blockIdx.y * 16;
  wmma_strip<3>(xcv, w, DI_, NDBC, DI_, row0, 0, acc);
  TILE_STORE_IDX();
#pragma unroll
  for (int t = 0; t < 3; ++t)
#pragma unroll
    for (int r = 0; r < 8; ++r)
      dbc[(size_t)(row0 + half * 8 + r) * NDBC + t * 16 + l16] = acc[t][r];
}

// ---------------- 4) dt = softplus(dbc[:, :16] @ dt_proj_w + b) ------------
__global__ void k_dt(const float* __restrict__ dbc,
                     const float* __restrict__ w,
                     const float* __restrict__ bias,
                     float* __restrict__ dt) {
  v8f acc[4] = {};
  const int row0 = blockIdx.y * 16;
  const int col0 = blockIdx.x * 64;
  wmma_strip<4>(dbc, w, NDBC, DI_, DTR_, row0, col0, acc);
  TILE_STORE_IDX();
#pragma unroll
  for (int t = 0; t < 4; ++t)
#pragma unroll
    for (int r = 0; r < 8; ++r) {
      const int col = col0 + t * 16 + l16;
      float v = acc[t][r] + bias[col];
      v = (v > 20.f) ? v : log1pf(expf(v));     // softplus
      dt[(size_t)(row0 + half * 8 + r) * DI_ + col] = v;
    }
}

// ---------------- 5) chunked selective scan --------------------------------
// phase 1: per (b, chunk, d): running h from 0 and cumulative a-product.
__global__ void k_scan_phase1(const float* __restrict__ dbc,
                              const float* __restrict__ dt,
                              const float* __restrict__ xcv,
                              const float* __restrict__ A_log,
                              float* __restrict__ apbuf,
                              float* __restrict__ hbuf) {
  __shared__ float Bs[CS_ * DS_];
  const int c = blockIdx.x, b = blockIdx.y;
  const int d = threadIdx.x;                    // 512 threads
  for (int i = threadIdx.x; i < CS_ * DS_; i += blockDim.x) {
    const int s = i / DS_, n = i % DS_;
    Bs[i] = dbc[(size_t)(b * S_ + c * CS_ + s) * NDBC + DTR_ + n];
  }
  __syncthreads();
  float An[DS_], h[DS_], ap[DS_];
#pragma unroll
  for (int n = 0; n < DS_; ++n) {
    An[n] = -expf(A_log[d * DS_ + n]);
    h[n]  = 0.f;
    ap[n] = 1.f;
  }
  for (int s = 0; s < CS_; ++s) {
    const size_t ridx = (size_t)(b * S_ + c * CS_ + s) * DI_ + d;
    const float dtv = dt[ridx];
    const float bx  = dtv * xcv[ridx];
#pragma unroll
    for (int n = 0; n < DS_; ++n) {
      const float a = expf(dtv * An[n]);
      ap[n] *= a;
      h[n] = a * h[n] + bx * Bs[s * DS_ + n];
    }
  }
  const size_t base = ((size_t)(b * NC_ + c) * DI_ + d) * DS_;
#pragma unroll
  for (int n = 0; n < DS_; ++n) {
    apbuf[base + n] = ap[n];
    hbuf[base + n]  = h[n];
  }
}

// phase 2: tiny cross-chunk scan; leaves each chunk's INITIAL state in hbuf.
__global__ void k_scan_phase2(const float* __restrict__ apbuf,
                              float* __restrict__ hbuf) {
  const int t = blockIdx.x * blockDim.x + threadIdx.x;
  if (t >= B_ * DI_ * DS_) return;
  const int n = t % DS_;
  const int d = (t / DS_) % DI_;
  const int b = t / (DS_ * DI_);
  float h = 0.f;
  for (int c = 0; c < NC_; ++c) {
    const size_t idx = ((size_t)(b * NC_ + c) * DI_ + d) * DS_ + n;
    const float a  = apbuf[idx];
    const float hc = hbuf[idx];
    hbuf[idx] = h;                               // chunk-entry state
    h = a * h + hc;
  }
}

// phase 3: replay each chunk from corrected state; y = h.C + D*x, gated silu(z)
__global__ void k_scan_phase3(const float* __restrict__ dbc,
                              const float* __restrict__ dt,
                              const float* __restrict__ xcv,
                              const float* __restrict__ xz,
                              const float* __restrict__ A_log,
                              const float* __restrict__ Dskip,
                              const float* __restrict__ hbuf,
                              float* __restrict__ y) {
  __shared__ float Bs[CS_ * DS_];
  __shared__ float Cs[CS_ * DS_];
  const int c = blockIdx.x, b = blockIdx.y;
  const int d = threadIdx.x;
  for (int i = threadIdx.x; i < CS_ * DS_; i += blockDim.x) {
    const int s = i / DS_, n = i % DS_;
    const size_t rb = (size_t)(b * S_ + c * CS_ + s) * NDBC;
    Bs[i] = dbc[rb + DTR_ + n];
    Cs[i] = dbc[rb + DTR_ + DS_ + n];
  }
  __syncthreads();
  float An[DS_], h[DS_];
  const size_t base = ((size_t)(b * NC_ + c) * DI_ + d) * DS_;
#pragma unroll
  for (int n = 0; n < DS_; ++n) {
    An[n] = -expf(A_log[d * DS_ + n]);
    h[n]  = hbuf[base + n];
  }
  const float dsk = Dskip[d];
  for (int s = 0; s < CS_; ++s) {
    const size_t ridx = (size_t)(b * S_ + c * CS_ + s) * DI_ + d;
    const float dtv = dt[ridx];
    const float xv  = xcv[ridx];
    const float bx  = dtv * xv;
    float acc = dsk * xv;
#pragma unroll
    for (int n = 0; n < DS_; ++n) {
      const float a = expf(dtv * An[n]);
      h[n] = a * h[n] + bx * Bs[s * DS_ + n];
      acc += h[n] * Cs[s * DS_ + n];
    }
    const float zv = xz[(size_t)(b * S_ + c * CS_ + s) * (2 * DI_) + DI_ + d];
    y[ridx] = acc * silu_(zv);
  }
}

// ---------------- 6) m = y @ out_proj_w ------------------------------------
__global__ void k_out_proj(const float* __restrict__ y,
                           const float* __restrict__ w,
                           float* __restrict__ m) {
  v8f acc[4] = {};
  const int row0 = blockIdx.y * 16;
  const int col0 = blockIdx.x * 64;
  wmma_strip<4>(y, w, DI_, D_, DI_, row0, col0, acc);
  TILE_STORE_IDX();
#pragma unroll
  for (int t = 0; t < 4; ++t)
#pragma unroll
    for (int r = 0; r < 8; ++r)
      m[(size_t)(row0 + half * 8 + r) * D_ + col0 + t * 16 + l16] = acc[t][r];
}

// ---------------- 7) h1 = elu(m @ w1 + b1) ---------------------------------
__global__ void k_mlp1(const float* __restrict__ m,
                       const float* __restrict__ w1,
                       const float* __restrict__ b1,
                       float* __restrict__ h1) {
  v8f acc[4] = {};
  const int row0 = blockIdx.y * 16;
  const int col0 = blockIdx.x * 64;
  wmma_strip<4>(m, w1, D_, W_, D_, row0, col0, acc);
  TILE_STORE_IDX();
#pragma unroll
  for (int t = 0; t < 4; ++t)
#pragma unroll
    for (int r = 0; r < 8; ++r) {
      const int col = col0 + t * 16 + l16;
      h1[(size_t)(row0 + half * 8 + r) * W_ + col] = elu_(acc[t][r] + b1[col]);
    }
}

// ---------------- 8) u = m + elu(h1 @ w2 + b2) -----------------------------
__global__ void k_mlp2(const float* __restrict__ h1,
                       const float* __restrict__ w2,
                       const float* __restrict__ b2,
                       const float* __restrict__ m,
                       float* __restrict__ u) {
  v8f acc[4] = {};
  const int row0 = blockIdx.y * 16;
  const int col0 = blockIdx.x * 64;
  wmma_strip<4>(h1, w2, W_, D_, W_, row0, col0, acc);
  TILE_STORE_IDX();
#pragma unroll
  for (int t = 0; t < 4; ++t)
#pragma unroll
    for (int r = 0; r < 8; ++r) {
      const int col = col0 + t * 16 + l16;
      const size_t o = (size_t)(row0 + half * 8 + r) * D_ + col;
      u[o] = m[o] + elu_(acc[t][r] + b2[col]);
    }
}

// ---------------- 9) LayerNorm + mask --------------------------------------
__global__ void k_ln_mask(const float* __restrict__ u,
                          const float* __restrict__ g,
                          const float* __restrict__ bb,
                          const unsigned char* __restrict__ mask,
                          float* __restrict__ out) {
  const int row = blockIdx.x;     // 0..8191
  const int col = threadIdx.x;    // 0..255
  __shared__ float red[2][8];     // 8 wave32 partial sums
  const float v = u[(size_t)row * D_ + col];
  float s1 = v, s2 = v * v;
#pragma unroll
  for (int off = 16; off > 0; off >>= 1) {
    s1 += __shfl_down(s1, off, 32);
    s2 += __shfl_down(s2, off, 32);
  }
  const int wave = col >> 5, lane = col & 31;
  if (lane == 0) { red[0][wave] = s1; red[1][wave] = s2; }
  __syncthreads();
  float t1 = 0.f, t2 = 0.f;
#pragma unroll
  for (int i = 0; i < 8; ++i) { t1 += red[0][i]; t2 += red[1][i]; }
  const float mu  = t1 * (1.f / D_);
  const float var = t2 * (1.f / D_) - mu * mu;
  const float rs  = rsqrtf(var + 1e-5f);
  const float keep = mask[row] ? 0.f : 1.f;
  out[(size_t)row * D_ + col] = ((v - mu) * rs * g[col] + bb[col]) * keep;
}

// ---------------- launch ---------------------------------------------------
extern "C" void kernel_launch(void* const* d_in, const int* in_sizes, int n_in,
                              void* d_out, int out_size, void* d_ws,
                              size_t ws_size, hipStream_t stream) {
  const float* x       = (const float*)d_in[0];
  const float* in_w    = (const float*)d_in[1];
  const float* conv_w  = (const float*)d_in[2];
  const float* conv_b  = (const float*)d_in[3];
  const float* xproj_w = (const float*)d_in[4];
  const float* dt_w    = (const float*)d_in[5];
  const float* dt_b    = (const float*)d_in[6];
  const float* A_log   = (const float*)d_in[7];
  const float* Dskip   = (const float*)d_in[8];
  const float* out_w   = (const float*)d_in[9];
  const float* ln_g    = (const float*)d_in[10];
  const float* ln_b    = (const float*)d_in[11];
  const float* w1      = (const float*)d_in[12];
  const float* b1      = (const float*)d_in[13];
  const float* w2      = (const float*)d_in[14];
  const float* b2      = (const float*)d_in[15];
  const unsigned char* mask = (const unsigned char*)d_in[16];
  float* out = (float*)d_out;

  float* p = (float*)d_ws;
  float* xz    = p; p += (size_t)NROW * 2 * DI_;          // 32 MB
  float* xcv   = p; p += (size_t)NROW * DI_;              // 16 MB
  float* dbc   = p; p += (size_t)NROW * NDBC;             // 1.5 MB
  float* dtb   = p; p += (size_t)NROW * DI_;              // 16 MB
  float* apbuf = p; p += (size_t)B_ * NC_ * DI_ * DS_;    // 2 MB
  float* hbuf  = p; p += (size_t)B_ * NC_ * DI_ * DS_;    // 2 MB
  float* yb    = p; p += (size_t)NROW * DI_;              // 16 MB
  float* mb    = p; p += (size_t)NROW * D_;               // 8 MB
  float* h1b   = p; p += (size_t)NROW * D_;               // 8 MB
  float* ub    = p; p += (size_t)NROW * D_;               // 8 MB

  k_in_proj   <<<dim3(2 * DI_ / 64, NROW / 16), 32, 0, stream>>>(x, in_w, xz);
  k_conv_silu <<<(NROW * DI_ + 255) / 256, 256, 0, stream>>>(xz, conv_w, conv_b, xcv);
  k_x_proj    <<<dim3(1, NROW / 16), 32, 0, stream>>>(xcv, xproj_w, dbc);
  k_dt        <<<dim3(DI_ / 64, NROW / 16), 32, 0, stream>>>(dbc, dt_w, dt_b, dtb);
  k_scan_phase1<<<dim3(NC_, B_), DI_, 0, stream>>>(dbc, dtb, xcv, A_log, apbuf, hbuf);
  k_scan_phase2<<<(B_ * DI_ * DS_ + 255) / 256, 256, 0, stream>>>(apbuf, hbuf);
  k_scan_phase3<<<dim3(NC_, B_), DI_, 0, stream>>>(dbc, dtb, xcv, xz, A_log, Dskip, hbuf, yb);
  k_out_proj  <<<dim3(D_ / 64, NROW / 16), 32, 0, stream>>>(yb, out_w, mb);
  k_mlp1      <<<dim3(W_ / 64, NROW / 16), 32, 0, stream>>>(mb, w1, b1, h1b);
  k_mlp2      <<<dim3(D_ / 64, NROW / 16), 32, 0, stream>>>(h1b, w2, b2, mb, ub);
  k_ln_mask   <<<NROW, D_, 0, stream>>>(ub, ln_g, ln_b, mask, out);
}